// LabelRelevance_63848983822523
// MI455X (gfx1250) — compile-verified
//
#include <hip/hip_runtime.h>

// LabelRelevance (AUC-style pairwise loss) for MI455X / gfx1250.
//
// Bandwidth-bound: 134 MB of reads -> ~5.8us floor at 23.3 TB/s.
// V_WMMA_F32_16X16X4_F32 with an all-ones B matrix is the 16-row reduction
// engine (D[m][*] = running row sum). Lane L supplies row L%16 values
// (A-layout requirement); each b128 load feeds two WMMAs. Three quantities
// (sum exp(o) over t==0, sum exp(-o) over t==1, count of t==1) use two
// interleaved accumulators each to halve the WMMA RAW chain.
//
// Round-1 asm showed the scheduler pipelines loads only 1 span deep, so this
// version doubles wave-level MLP: each 16-row tile is split across 2 waves
// (256 columns each, 4096 waves total), with the per-row partial sums merged
// through LDS before the nonlinear s_neg*s_pos/num step. Prefetch cadence is
// now one global_prefetch_b8 per 128B line (every 4 spans), ~512B ahead.

typedef float v2f __attribute__((ext_vector_type(2)));
typedef float v4f __attribute__((ext_vector_type(4)));
typedef float v8f __attribute__((ext_vector_type(8)));
typedef int   v4i __attribute__((ext_vector_type(4)));

#define CNUM 512
#define KHALF 256   // columns handled per wave

__global__ __launch_bounds__(256) void auc_wmma_kernel(
    const float* __restrict__ outp,
    const int*   __restrict__ tgtp,
    float*       __restrict__ blockPartials)
{
    __shared__ float sN[8][16];   // per-wave row sums: sum exp(o), t==0
    __shared__ float sP[8][16];   // per-wave row sums: sum exp(-o), t==1
    __shared__ float sC[8][16];   // per-wave row counts of t==1
    __shared__ float red[8];

    const int lane  = threadIdx.x & 31;
    const int wv    = threadIdx.x >> 5;     // 0..7
    const int tl    = wv >> 1;              // local tile 0..3
    const int khalf = wv & 1;               // which K half this wave streams
    const int tile  = blockIdx.x * 4 + tl;  // 16 rows per tile
    const int row   = tile * 16 + (lane & 15);   // lane L feeds row L%16
    const int hi    = lane >> 4;                 // lanes 16..31 -> K={2,3} slots

    const float* orow = outp + (size_t)row * CNUM + khalf * KHALF + 4 * hi;
    const int*   trow = tgtp + (size_t)row * CNUM + khalf * KHALF + 4 * hi;

    v8f nA = {}, nB = {};
    v8f pA = {}, pB = {};
    v8f cA = {}, cB = {};
    v2f ones; ones.x = 1.0f; ones.y = 1.0f;   // B(4x16)=ones: layout-agnostic

#pragma unroll 8
    for (int s = 0; s < KHALF / 8; ++s) {        // 32 spans of 8 columns
        v4f o = *(const v4f*)(orow + s * 8);     // global_load_b128
        v4i t = *(const v4i*)(trow + s * 8);     // global_load_b128

        if ((s & 3) == 0) {   // one prefetch per 128B line, ~512B ahead
            __builtin_prefetch(orow + s * 8 + 128, 0, 0);
            __builtin_prefetch(trow + s * 8 + 128, 0, 0);
        }

        float e0 = __expf(o.x), e1 = __expf(o.y), e2 = __expf(o.z), e3 = __expf(o.w);
        float g0 = __expf(-o.x), g1 = __expf(-o.y), g2 = __expf(-o.z), g3 = __expf(-o.w);

        v2f an0, an1, ap0, ap1, ac0, ac1;
        an0.x = t.x ? 0.0f : e0;   an0.y = t.y ? 0.0f : e1;
        an1.x = t.z ? 0.0f : e2;   an1.y = t.w ? 0.0f : e3;
        ap0.x = t.x ? g0 : 0.0f;   ap0.y = t.y ? g1 : 0.0f;
        ap1.x = t.z ? g2 : 0.0f;   ap1.y = t.w ? g3 : 0.0f;
        ac0.x = t.x ? 1.0f : 0.0f; ac0.y = t.y ? 1.0f : 0.0f;
        ac1.x = t.z ? 1.0f : 0.0f; ac1.y = t.w ? 1.0f : 0.0f;

        // D = A x ones + C : every column of D is the running row sum.
        nA = __builtin_amdgcn_wmma_f32_16x16x4_f32(false, an0, false, ones, (short)0, nA, false, false);
        nB = __builtin_amdgcn_wmma_f32_16x16x4_f32(false, an1, false, ones, (short)0, nB, false, false);
        pA = __builtin_amdgcn_wmma_f32_16x16x4_f32(false, ap0, false, ones, (short)0, pA, false, false);
        pB = __builtin_amdgcn_wmma_f32_16x16x4_f32(false, ap1, false, ones, (short)0, pB, false, false);
        cA = __builtin_amdgcn_wmma_f32_16x16x4_f32(false, ac0, false, ones, (short)0, cA, false, false);
        cB = __builtin_amdgcn_wmma_f32_16x16x4_f32(false, ac1, false, ones, (short)0, cB, false, false);
    }

    // C/D layout: accumulator VGPR j holds rowsum(j) in lanes 0..15 and
    // rowsum(j+8) in lanes 16..31, replicated across N. Lanes 0 and 16
    // publish this wave's 16 row-partials to LDS.
    float rn[8], rp[8], rc[8];
#pragma unroll
    for (int j = 0; j < 8; ++j) {
        rn[j] = nA[j] + nB[j];
        rp[j] = pA[j] + pB[j];
        rc[j] = cA[j] + cB[j];
    }
    if ((lane & 15) == 0) {          // lanes 0 (rows 0..7) and 16 (rows 8..15)
        const int base = hi * 8;
#pragma unroll
        for (int j = 0; j < 8; ++j) {
            sN[wv][base + j] = rn[j];
            sP[wv][base + j] = rp[j];
            sC[wv][base + j] = rc[j];
        }
    }
    __syncthreads();

    // Merge the two K-halves per tile, apply the nonlinear per-row term.
    float term = 0.0f;
    if (threadIdx.x < 64) {                       // 4 tiles x 16 rows
        const int t  = threadIdx.x >> 4;          // local tile
        const int r  = threadIdx.x & 15;          // row in tile
        const int wA = t * 2, wB = wA + 1;
        float sn = sN[wA][r] + sN[wB][r];
        float sp = sP[wA][r] + sP[wB][r];
        float np = sC[wA][r] + sC[wB][r];         // exact int in f32 (<=512)
        float den = np * ((float)CNUM - np);      // |pos| * |neg|
        den = (den == 0.0f) ? (float)CNUM : den;
        term = sn * sp / den;
    }
#pragma unroll
    for (int off = 16; off > 0; off >>= 1) term += __shfl_down(term, off);
    if ((threadIdx.x & 31) == 0) red[threadIdx.x >> 5] = term;
    __syncthreads();
    if (threadIdx.x == 0) {
        float b = 0.0f;
#pragma unroll
        for (int w = 0; w < 8; ++w) b += red[w];
        blockPartials[blockIdx.x] = b;            // no global atomics
    }
}

__global__ __launch_bounds__(256) void final_reduce_kernel(
    const float* __restrict__ ws, float* __restrict__ loss, int n)
{
    __shared__ float sh[8];
    float v = 0.0f;
    for (int i = threadIdx.x; i < n; i += 256) v += ws[i];
#pragma unroll
    for (int off = 16; off > 0; off >>= 1) v += __shfl_down(v, off);
    if ((threadIdx.x & 31) == 0) sh[threadIdx.x >> 5] = v;
    __syncthreads();
    if (threadIdx.x == 0) {
        float r = 0.0f;
#pragma unroll
        for (int w = 0; w < 8; ++w) r += sh[w];
        *loss = r;
    }
}

extern "C" void kernel_launch(void* const* d_in, const int* in_sizes, int n_in,
                              void* d_out, int out_size, void* d_ws, size_t ws_size,
                              hipStream_t stream)
{
    const float* outp = (const float*)d_in[0];
    const int*   tgtp = (const int*)d_in[1];
    float*       loss = (float*)d_out;
    float*       ws   = (float*)d_ws;

    const int rows   = in_sizes[0] / CNUM;   // 32768
    const int tiles  = rows / 16;            // 2048 tiles of 16 rows
    const int blocks = tiles / 4;            // 512 blocks: 4 tiles x 2 K-half waves

    auc_wmma_kernel<<<blocks, 256, 0, stream>>>(outp, tgtp, ws);
    final_reduce_kernel<<<1, 256, 0, stream>>>(ws, loss, blocks);
}